// chamfer_distance_59923383714072
// MI455X (gfx1250) — compile-verified
//
#include <hip/hip_runtime.h>

typedef float v2f __attribute__((ext_vector_type(2)));
typedef float v8f __attribute__((ext_vector_type(8)));

#define H_DIM 64
#define W_DIM 128
#define NPTS 8192          // H*W
#define NPAIR 8            // B*S
#define NPD 16             // pair-directions
#define NBLK 64            // row blocks per pair-direction (8192 / 128)
#define BIGF 1.0e9f
#define PI_F 3.14159265358979323846f

// ---------------------------------------------------------------------------
// Kernel 1: spherical back-projection + norms + mask penalties.
//   colO[p][n] = {x, y, z, |p|^2 + BIG*(1-mo)}   (rv points)
//   colT[p][n] = {x, y, z, |q|^2 + BIG*(1-mt)}   (target points)
// ---------------------------------------------------------------------------
__global__ void __launch_bounds__(256)
precompute_kernel(const float* __restrict__ rv, const float* __restrict__ tgt,
                  float4* __restrict__ colO, float4* __restrict__ colT,
                  float* __restrict__ maskO, float* __restrict__ maskT) {
  const int p = blockIdx.y;                      // pair index b*S+s
  const int n = blockIdx.x * blockDim.x + threadIdx.x;  // point index h*W+w
  const int h = n >> 7;
  const int w = n & (W_DIM - 1);

  const float FOV_UP = 3.0f * PI_F / 180.0f;
  const float FOV_DOWN = -25.0f * PI_F / 180.0f;
  const float fov = FOV_UP - FOV_DOWN;

  // range-view point
  const float r = rv[p * NPTS + n];
  const float pitch = (1.0f - ((float)h + 0.5f) / (float)H_DIM) * fov + FOV_DOWN;
  const float yaw = -((((float)w + 0.5f) / (float)W_DIM) * 2.0f - 1.0f) * PI_F;
  float sp, cp, sy, cy;
  __sincosf(pitch, &sp, &cp);
  __sincosf(yaw, &sy, &cy);
  const float x = r * cp * cy;
  const float y = r * cp * sy;
  const float z = r * sp;
  const float mo = (r > 0.0f) ? 1.0f : 0.0f;
  const float no = x * x + y * y + z * z;
  colO[p * NPTS + n] = make_float4(x, y, z, no + BIGF * (1.0f - mo));
  maskO[p * NPTS + n] = mo;

  // target point (channels 1..3), mask from channel 0
  const float* tb = tgt + (size_t)p * 4 * NPTS;
  const float t0 = tb[0 * NPTS + n];
  const float tx = tb[1 * NPTS + n];
  const float ty = tb[2 * NPTS + n];
  const float tz = tb[3 * NPTS + n];
  const float mt = (t0 > 0.0f) ? 1.0f : 0.0f;
  const float nt = tx * tx + ty * ty + tz * tz;
  colT[p * NPTS + n] = make_float4(tx, ty, tz, nt + BIGF * (1.0f - mt));
  maskT[p * NPTS + n] = mt;
}

// ---------------------------------------------------------------------------
// Kernel 2: for pair-direction pd, compute sum_i mask_i * (a_i + min_j t[i][j])
// using V_WMMA_F32_16X16X4_F32 tiles.  grid = (row-block 64, pd 16), 256 thr.
// Each wave owns 16 rows; column set streamed through 32KB LDS chunks.
// ---------------------------------------------------------------------------
__global__ void __launch_bounds__(256)
chamfer_min_kernel(const float4* __restrict__ colO, const float4* __restrict__ colT,
                   const float* __restrict__ maskO, const float* __restrict__ maskT,
                   float* __restrict__ partial /* [NPD][NBLK] */) {
  __shared__ float4 ldsQ[2048];   // 32 KB chunk of the column point set
  __shared__ float wsum[16];

  const int pd = blockIdx.y;
  const int p = pd >> 1;
  const int dir = pd & 1;
  const float4* __restrict__ colR = dir ? (colT + p * NPTS) : (colO + p * NPTS);
  const float4* __restrict__ colQ = dir ? (colO + p * NPTS) : (colT + p * NPTS);
  const float* __restrict__ mR    = dir ? (maskT + p * NPTS) : (maskO + p * NPTS);

  const int tid = threadIdx.x;
  const int lane = tid & 31;
  const int wave = tid >> 5;
  const int hl = lane & 15;
  const bool hi = lane >= 16;

  // A matrix (16x4 fp32): lane m<16 holds K={0,1}, lane m>=16 holds K={2,3}.
  // Row i: [-2x, -2y, -2z, 1]
  const int i0 = blockIdx.x * 128 + wave * 16;
  const float4 pr = colR[i0 + hl];
  v2f a;
  a.x = hi ? (-2.0f * pr.z) : (-2.0f * pr.x);
  a.y = hi ? 1.0f           : (-2.0f * pr.y);

  v8f minacc;
#pragma unroll
  for (int k = 0; k < 8; ++k) minacc[k] = 3.0e38f;
  const v8f czero = {};

  for (int c = 0; c < 4; ++c) {
    const float4* __restrict__ src = colQ + c * 2048;
#pragma unroll
    for (int k = 0; k < 8; ++k) ldsQ[tid + k * 256] = src[tid + k * 256];
    __syncthreads();

#pragma unroll 4
    for (int jt = 0; jt < 128; ++jt) {
      // B matrix (4x16 fp32): lane n<16 holds K={0,1} = (x,y),
      // lane n>=16 holds K={2,3} = (z, |q|^2 + BIG*(1-m)).
      const float2* bp = (const float2*)&ldsQ[jt * 16 + hl];
      const float2 bv = bp[hi ? 1 : 0];
      v2f b;
      b.x = bv.x;
      b.y = bv.y;
      // t[i][j] = -2*dot(p_i,q_j) + |q_j|^2 + BIG*(1-m_j)  for a 16x16 tile
      v8f t = __builtin_amdgcn_wmma_f32_16x16x4_f32(
          false, a, false, b, (short)0, czero, false, false);
#pragma unroll
      for (int k = 0; k < 8; ++k) minacc[k] = fminf(minacc[k], t[k]);
    }
    __syncthreads();
  }

  // Column-min finish: C/D tile has N = lane&15, rows M=v (lanes 0-15) and
  // M=v+8 (lanes 16-31).  Min-reduce across each 16-lane half.
#pragma unroll
  for (int m = 1; m < 16; m <<= 1) {
#pragma unroll
    for (int k = 0; k < 8; ++k)
      minacc[k] = fminf(minacc[k], __shfl_xor(minacc[k], m, 16));
  }

  if (hl == 0) {
    const int rbase = i0 + (hi ? 8 : 0);
    float local = 0.0f;
#pragma unroll
    for (int k = 0; k < 8; ++k) {
      const float4 q = colR[rbase + k];
      const float a2 = q.x * q.x + q.y * q.y + q.z * q.z;  // a_i = |p_i|^2
      local += (a2 + minacc[k]) * mR[rbase + k];
    }
    wsum[wave * 2 + (hi ? 1 : 0)] = local;
  }
  __syncthreads();
  if (tid == 0) {
    float s = 0.0f;
#pragma unroll
    for (int k = 0; k < 16; ++k) s += wsum[k];
    partial[pd * NBLK + blockIdx.x] = s;
  }
}

// ---------------------------------------------------------------------------
// Kernel 3: per pair-direction, sum the 64 block partials + the row-mask sum
// (deterministic tree reductions) -> val[pd] = sum / max(mask_sum, 1)
// ---------------------------------------------------------------------------
__global__ void __launch_bounds__(256)
reduce_pd_kernel(const float* __restrict__ partial,
                 const float* __restrict__ maskO, const float* __restrict__ maskT,
                 float* __restrict__ val) {
  __shared__ float sdat[256];
  __shared__ float smask[256];
  const int pd = blockIdx.x;
  const int p = pd >> 1;
  const int dir = pd & 1;
  const float* __restrict__ mR = dir ? (maskT + p * NPTS) : (maskO + p * NPTS);
  const int t = threadIdx.x;

  float s = (t < NBLK) ? partial[pd * NBLK + t] : 0.0f;
  float ms = 0.0f;
  for (int i = t; i < NPTS; i += 256) ms += mR[i];
  sdat[t] = s;
  smask[t] = ms;
  __syncthreads();
  for (int off = 128; off > 0; off >>= 1) {
    if (t < off) {
      sdat[t] += sdat[t + off];
      smask[t] += smask[t + off];
    }
    __syncthreads();
  }
  if (t == 0) val[pd] = sdat[0] / fmaxf(smask[0], 1.0f);
}

// ---------------------------------------------------------------------------
// Kernel 4: out[0:4] = per_step[s], out[4:12] = tensor[s][b] (row-major)
// pair value(p=b*S+s) = val[2p] + val[2p+1]
// ---------------------------------------------------------------------------
__global__ void assemble_kernel(const float* __restrict__ val, float* __restrict__ out) {
  const int t = threadIdx.x;
  if (t < 4) {
    const int s = t;
    const float pv0 = val[(0 * 4 + s) * 2] + val[(0 * 4 + s) * 2 + 1];
    const float pv1 = val[(1 * 4 + s) * 2] + val[(1 * 4 + s) * 2 + 1];
    out[s] = 0.5f * (pv0 + pv1);
  } else if (t < 12) {
    const int idx = t - 4;
    const int s = idx >> 1;
    const int b = idx & 1;
    out[t] = val[(b * 4 + s) * 2] + val[(b * 4 + s) * 2 + 1];
  }
}

// ---------------------------------------------------------------------------
extern "C" void kernel_launch(void* const* d_in, const int* in_sizes, int n_in,
                              void* d_out, int out_size, void* d_ws, size_t ws_size,
                              hipStream_t stream) {
  const float* rv = (const float*)d_in[0];
  const float* tgt = (const float*)d_in[1];
  // d_in[2] = mos_label (unused by reference output), d_in[3] = n_samples (-1)

  char* ws = (char*)d_ws;
  size_t off = 0;
  float4* colO = (float4*)(ws + off); off += (size_t)NPAIR * NPTS * sizeof(float4);
  float4* colT = (float4*)(ws + off); off += (size_t)NPAIR * NPTS * sizeof(float4);
  float* maskO = (float*)(ws + off);  off += (size_t)NPAIR * NPTS * sizeof(float);
  float* maskT = (float*)(ws + off);  off += (size_t)NPAIR * NPTS * sizeof(float);
  float* partial = (float*)(ws + off); off += (size_t)NPD * NBLK * sizeof(float);
  float* val = (float*)(ws + off);     off += (size_t)NPD * sizeof(float);

  precompute_kernel<<<dim3(NPTS / 256, NPAIR), 256, 0, stream>>>(
      rv, tgt, colO, colT, maskO, maskT);
  chamfer_min_kernel<<<dim3(NBLK, NPD), 256, 0, stream>>>(
      colO, colT, maskO, maskT, partial);
  reduce_pd_kernel<<<NPD, 256, 0, stream>>>(partial, maskO, maskT, val);
  assemble_kernel<<<1, 32, 0, stream>>>(val, (float*)d_out);
}